// ICP_91319594647596
// MI455X (gfx1250) — compile-verified
//
#include <hip/hip_runtime.h>
#include <math.h>

#define BATCH     2
#define NPTS      4096
#define MPTS      4096
#define ICP_STEPS 8
#define ICP_TOL   1e-6f

typedef float v2f __attribute__((ext_vector_type(2)));
typedef float v8f __attribute__((ext_vector_type(8)));

// ------------------------- tiny utility kernels -------------------------

__global__ void icp_copy_kernel(const float* __restrict__ src, float* __restrict__ dst, int n) {
  int i = blockIdx.x * blockDim.x + threadIdx.x;
  if (i < n) dst[i] = src[i];
}

__global__ void icp_zero_kernel(float* __restrict__ p, int n) {
  int i = blockIdx.x * blockDim.x + threadIdx.x;
  if (i < n) p[i] = 0.0f;
}

// Build augmented target table TA[b][m] = {-2tx, -2ty, -2tz, |t|^2} (16B rows).
__global__ void icp_build_ta_kernel(const float* __restrict__ ptgt, float* __restrict__ ta) {
  int i = blockIdx.x * blockDim.x + threadIdx.x;
  if (i >= BATCH * MPTS) return;
  float x = ptgt[3 * i + 0], y = ptgt[3 * i + 1], z = ptgt[3 * i + 2];
  float4 r;
  r.x = -2.0f * x;
  r.y = -2.0f * y;
  r.z = -2.0f * z;
  r.w = x * x + y * y + z * z;
  ((float4*)ta)[i] = r;
}

// ------------------------- WMMA nearest-neighbor kernel -------------------------
// One wave owns a 16-source tile; loops over all 256 target tiles.
// D[m][n] = |t_m|^2 - 2 t_m . p_n  via V_WMMA_F32_16X16X4_F32 with augmented K=4.
// Accumulates per-batch sums: [Sum p (3)][Sum q (3)][Sum p_i q_j (9)][Sum dist (1)].

__global__ __launch_bounds__(256) void icp_nn_kernel(
    const float* __restrict__ pc, const float* __restrict__ ta,
    const float* __restrict__ ptgt, float* __restrict__ acc) {
  const int lane = threadIdx.x & 31;
  const int col  = lane & 15;
  const int hi   = lane >> 4;  // 0: K=0,1 half / targets v .. v+7 ; 1: K=2,3 / targets v+8..v+15
  const int wave = blockIdx.x * (blockDim.x >> 5) + (threadIdx.x >> 5);
  const int b    = wave >> 8;   // NPTS/16 = 256 source tiles per batch
  const int tile = wave & 255;
  const int s    = tile * 16 + col;

  // B operand: source column [px, py, pz, 1] split across lane halves (K=0,1 | K=2,3)
  const float* pp = pc + ((size_t)b * NPTS + s) * 3;
  float px = pp[0], py = pp[1], pz = pp[2];
  v2f bop;
  bop.x = hi ? pz : px;
  bop.y = hi ? 1.0f : py;

  const float* tab = ta + (size_t)b * MPTS * 4;
  float best = 3.0e38f;
  int   bidx = 0;
  const int hioff = hi ? 8 : 0;

  for (int tb = 0; tb < MPTS; tb += 16) {
    const float* rowp = tab + (size_t)(tb + col) * 4 + hi * 2;
    __builtin_prefetch(rowp + 256, 0, 3);  // 4 tiles ahead -> global_prefetch_b8
    v2f aop = *(const v2f*)rowp;           // {-2tx,-2ty} or {-2tz,|t|^2}
    v8f cz = {};
    v8f d = __builtin_amdgcn_wmma_f32_16x16x4_f32(
        false, aop, false, bop, (short)0, cz, false, false);
#pragma unroll
    for (int v = 0; v < 8; ++v) {
      float cand = d[v];
      int idx = tb + v + hioff;
      bool lt = cand < best;
      bidx = lt ? idx : bidx;
      best = lt ? cand : best;
    }
  }

  // lanes L and L+16 cover the two target halves of source col: combine
  float obest = __shfl_xor(best, 16, 32);
  int   oidx  = __shfl_xor(bidx, 16, 32);
  bool lt = obest < best;
  best = lt ? obest : best;
  bidx = lt ? oidx : bidx;

  float pn   = px * px + py * py + pz * pz;
  float dist = sqrtf(fmaxf(best + pn, 0.0f));

  const float* q = ptgt + ((size_t)b * MPTS + bidx) * 3;
  float qx = q[0], qy = q[1], qz = q[2];

  // Wave-level partial sums: only lanes 0..15 contribute (one per source point).
  float m = hi ? 0.0f : 1.0f;
  float vals[16] = {
      px, py, pz, qx, qy, qz,
      px * qx, px * qy, px * qz,
      py * qx, py * qy, py * qz,
      pz * qx, pz * qy, pz * qz,
      dist};
  float* ab = acc + b * 16;
#pragma unroll
  for (int k = 0; k < 16; ++k) {
    float v = vals[k] * m;
    v += __shfl_xor(v, 16, 32);
    v += __shfl_xor(v, 8, 32);
    v += __shfl_xor(v, 4, 32);
    v += __shfl_xor(v, 2, 32);
    v += __shfl_xor(v, 1, 32);
    if (lane == 0) atomicAdd(ab + k, v);
  }
}

// ------------------------- 3x3 Kabsch (device) -------------------------

__device__ inline float icp_det3(const float M[3][3]) {
  return M[0][0] * (M[1][1] * M[2][2] - M[1][2] * M[2][1]) -
         M[0][1] * (M[1][0] * M[2][2] - M[1][2] * M[2][0]) +
         M[0][2] * (M[1][0] * M[2][1] - M[1][1] * M[2][0]);
}

// q ~= R p + t with reflection fix (matches SVD-based Kabsch up to sign conventions
// that cancel in R).
__device__ void icp_kabsch_rt(const float Hm[3][3], const float pm[3], const float qm[3],
                              float R[3][3], float tv[3]) {
  // S = H^T H (symmetric PSD)
  float A[3][3];
  for (int i = 0; i < 3; ++i)
    for (int j = 0; j < 3; ++j) {
      float sum = 0.0f;
      for (int k = 0; k < 3; ++k) sum += Hm[k][i] * Hm[k][j];
      A[i][j] = sum;
    }
  float V[3][3] = {{1, 0, 0}, {0, 1, 0}, {0, 0, 1}};
  const int PR[3] = {0, 0, 1};
  const int QR[3] = {1, 2, 2};
  for (int sweep = 0; sweep < 10; ++sweep) {
    for (int r = 0; r < 3; ++r) {
      int p = PR[r], q = QR[r];
      float apq = A[p][q];
      if (fabsf(apq) > 1e-22f) {
        float tau = (A[q][q] - A[p][p]) / (2.0f * apq);
        float sg = (tau >= 0.0f) ? 1.0f : -1.0f;
        float tt = sg / (fabsf(tau) + sqrtf(1.0f + tau * tau));
        float c = 1.0f / sqrtf(1.0f + tt * tt);
        float sn = tt * c;
        for (int k = 0; k < 3; ++k) {
          float akp = A[k][p], akq = A[k][q];
          A[k][p] = c * akp - sn * akq;
          A[k][q] = sn * akp + c * akq;
        }
        for (int k = 0; k < 3; ++k) {
          float apk = A[p][k], aqk = A[q][k];
          A[p][k] = c * apk - sn * aqk;
          A[q][k] = sn * apk + c * aqk;
        }
        for (int k = 0; k < 3; ++k) {
          float vkp = V[k][p], vkq = V[k][q];
          V[k][p] = c * vkp - sn * vkq;
          V[k][q] = sn * vkp + c * vkq;
        }
      }
    }
  }
  float lam[3] = {A[0][0], A[1][1], A[2][2]};
  // sort eigenpairs descending (columns of V)
  for (int a0 = 0; a0 < 2; ++a0)
    for (int a1 = a0 + 1; a1 < 3; ++a1)
      if (lam[a1] > lam[a0]) {
        float l = lam[a0]; lam[a0] = lam[a1]; lam[a1] = l;
        for (int k = 0; k < 3; ++k) {
          float v = V[k][a0]; V[k][a0] = V[k][a1]; V[k][a1] = v;
        }
      }
  // U columns: u0 = normalize(H v0), u1 = normalize(H v1 - proj_u0), u2 = u0 x u1
  float U[3][3];
  float u0[3], u1[3];
  for (int i = 0; i < 3; ++i) {
    u0[i] = Hm[i][0] * V[0][0] + Hm[i][1] * V[1][0] + Hm[i][2] * V[2][0];
    u1[i] = Hm[i][0] * V[0][1] + Hm[i][1] * V[1][1] + Hm[i][2] * V[2][1];
  }
  float n0 = sqrtf(fmaxf(u0[0] * u0[0] + u0[1] * u0[1] + u0[2] * u0[2], 1e-30f));
  for (int i = 0; i < 3; ++i) u0[i] /= n0;
  float dot01 = u1[0] * u0[0] + u1[1] * u0[1] + u1[2] * u0[2];
  for (int i = 0; i < 3; ++i) u1[i] -= dot01 * u0[i];
  float n1 = sqrtf(fmaxf(u1[0] * u1[0] + u1[1] * u1[1] + u1[2] * u1[2], 1e-30f));
  for (int i = 0; i < 3; ++i) u1[i] /= n1;
  float u2[3] = {u0[1] * u1[2] - u0[2] * u1[1],
                 u0[2] * u1[0] - u0[0] * u1[2],
                 u0[0] * u1[1] - u0[1] * u1[0]};
  for (int i = 0; i < 3; ++i) {
    U[i][0] = u0[i]; U[i][1] = u1[i]; U[i][2] = u2[i];
  }
  float dd = icp_det3(V) * icp_det3(U);  // +-1 reflection fix
  for (int i = 0; i < 3; ++i)
    for (int j = 0; j < 3; ++j)
      R[i][j] = V[i][0] * U[j][0] + V[i][1] * U[j][1] + dd * V[i][2] * U[j][2];
  for (int i = 0; i < 3; ++i)
    tv[i] = qm[i] - (R[i][0] * pm[0] + R[i][1] * pm[1] + R[i][2] * pm[2]);
}

// ------------------------- per-step Kabsch + convergence -------------------------

__global__ void icp_kabsch_step_kernel(const float* __restrict__ acc,
                                       float* __restrict__ errv, int* __restrict__ done,
                                       float* __restrict__ Rt) {
  __shared__ float s_err[BATCH];
  int b = threadIdx.x;
  if (b < BATCH) {
    const float* a = acc + b * 16;
    const float invN = 1.0f / (float)NPTS;
    float pm[3], qm[3];
    for (int i = 0; i < 3; ++i) { pm[i] = a[i] * invN; qm[i] = a[3 + i] * invN; }
    float Hm[3][3];
    for (int i = 0; i < 3; ++i)
      for (int j = 0; j < 3; ++j)
        Hm[i][j] = a[6 + i * 3 + j] - (float)NPTS * pm[i] * qm[j];
    float R[3][3], tv[3];
    icp_kabsch_rt(Hm, pm, qm, R, tv);
    float* rt = Rt + b * 12;
    for (int i = 0; i < 3; ++i)
      for (int j = 0; j < 3; ++j) rt[i * 3 + j] = R[i][j];
    rt[9] = tv[0]; rt[10] = tv[1]; rt[11] = tv[2];
    s_err[b] = a[15] * invN;
  }
  __syncthreads();
  if (threadIdx.x == 0) {
    int conv = 1;
    for (int bb = 0; bb < BATCH; ++bb)
      if (!(fabsf(s_err[bb] - errv[bb]) < ICP_TOL)) conv = 0;
    int dn = (*done) || conv;
    if (!dn)
      for (int bb = 0; bb < BATCH; ++bb) errv[bb] = s_err[bb];
    *done = dn;
  }
}

__global__ void icp_transform_kernel(float* __restrict__ pc, const float* __restrict__ Rt,
                                     const int* __restrict__ done) {
  if (*done) return;
  int i = blockIdx.x * blockDim.x + threadIdx.x;
  if (i >= BATCH * NPTS) return;
  int b = i / NPTS;
  const float* rt = Rt + b * 12;
  float* p = pc + (size_t)i * 3;
  float x = p[0], y = p[1], z = p[2];
  p[0] = rt[0] * x + rt[1] * y + rt[2] * z + rt[9];
  p[1] = rt[3] * x + rt[4] * y + rt[5] * z + rt[10];
  p[2] = rt[6] * x + rt[7] * y + rt[8] * z + rt[11];
}

// ------------------------- final Kabsch psrc -> pc -------------------------

__global__ void icp_final_acc_kernel(const float* __restrict__ psrc,
                                     const float* __restrict__ pc,
                                     float* __restrict__ acc) {
  int i = blockIdx.x * blockDim.x + threadIdx.x;  // exactly BATCH*NPTS threads
  int b = i / NPTS;
  int lane = threadIdx.x & 31;
  const float* p = psrc + (size_t)i * 3;
  const float* q = pc + (size_t)i * 3;
  float px = p[0], py = p[1], pz = p[2];
  float qx = q[0], qy = q[1], qz = q[2];
  float vals[15] = {
      px, py, pz, qx, qy, qz,
      px * qx, px * qy, px * qz,
      py * qx, py * qy, py * qz,
      pz * qx, pz * qy, pz * qz};
  float* ab = acc + b * 16;
#pragma unroll
  for (int k = 0; k < 15; ++k) {
    float v = vals[k];
    v += __shfl_xor(v, 16, 32);
    v += __shfl_xor(v, 8, 32);
    v += __shfl_xor(v, 4, 32);
    v += __shfl_xor(v, 2, 32);
    v += __shfl_xor(v, 1, 32);
    if (lane == 0) atomicAdd(ab + k, v);
  }
}

__global__ void icp_final_out_kernel(const float* __restrict__ acc, float* __restrict__ out) {
  int b = threadIdx.x;
  if (b >= BATCH) return;
  const float* a = acc + b * 16;
  const float invN = 1.0f / (float)NPTS;
  float pm[3], qm[3];
  for (int i = 0; i < 3; ++i) { pm[i] = a[i] * invN; qm[i] = a[3 + i] * invN; }
  float Hm[3][3];
  for (int i = 0; i < 3; ++i)
    for (int j = 0; j < 3; ++j)
      Hm[i][j] = a[6 + i * 3 + j] - (float)NPTS * pm[i] * qm[j];
  float R[3][3], tv[3];
  icp_kabsch_rt(Hm, pm, qm, R, tv);
  float r00 = R[0][0], r11 = R[1][1], r22 = R[2][2];
  float qw = 0.5f * sqrtf(fmaxf(1.0f + r00 + r11 + r22, 1e-12f));
  float qx = 0.5f * sqrtf(fmaxf(1.0f + r00 - r11 - r22, 1e-12f));
  float qy = 0.5f * sqrtf(fmaxf(1.0f - r00 + r11 - r22, 1e-12f));
  float qz = 0.5f * sqrtf(fmaxf(1.0f - r00 - r11 + r22, 1e-12f));
  qx = (R[2][1] - R[1][2] >= 0.0f) ? qx : -qx;
  qy = (R[0][2] - R[2][0] >= 0.0f) ? qy : -qy;
  qz = (R[1][0] - R[0][1] >= 0.0f) ? qz : -qz;
  float* o = out + b * 7;
  o[0] = tv[0]; o[1] = tv[1]; o[2] = tv[2];
  o[3] = qx; o[4] = qy; o[5] = qz; o[6] = qw;
}

// ------------------------- host-side orchestration -------------------------

extern "C" void kernel_launch(void* const* d_in, const int* in_sizes, int n_in,
                              void* d_out, int out_size, void* d_ws, size_t ws_size,
                              hipStream_t stream) {
  (void)in_sizes; (void)n_in; (void)out_size; (void)ws_size;
  const float* psrc = (const float*)d_in[0];
  const float* ptgt = (const float*)d_in[1];
  float* out = (float*)d_out;
  float* ws = (float*)d_ws;

  // workspace layout (floats)
  float* pc   = ws;                 // BATCH*NPTS*3  = 24576
  float* ta   = ws + 24576;         // BATCH*MPTS*4  = 32768
  float* acc  = ws + 57344;         // BATCH*16      = 32
  float* errv = ws + 57376;         // BATCH         = 2
  int*   done = (int*)(ws + 57378); // 1
  float* Rt   = ws + 57380;         // BATCH*12      = 24

  const int npc = BATCH * NPTS * 3;
  icp_copy_kernel<<<(npc + 255) / 256, 256, 0, stream>>>(psrc, pc, npc);
  icp_build_ta_kernel<<<(BATCH * MPTS + 255) / 256, 256, 0, stream>>>(ptgt, ta);
  icp_zero_kernel<<<1, 32, 0, stream>>>(errv, 4);  // err[2] + done + pad

  const int nn_waves  = BATCH * (NPTS / 16);             // 512
  const int nn_blocks = nn_waves / 8;                    // 256 thr = 8 waves/block

  for (int step = 0; step < ICP_STEPS; ++step) {
    icp_zero_kernel<<<1, 32, 0, stream>>>(acc, BATCH * 16);
    icp_nn_kernel<<<nn_blocks, 256, 0, stream>>>(pc, ta, ptgt, acc);
    icp_kabsch_step_kernel<<<1, 64, 0, stream>>>(acc, errv, done, Rt);
    icp_transform_kernel<<<(BATCH * NPTS + 255) / 256, 256, 0, stream>>>(pc, Rt, done);
  }

  icp_zero_kernel<<<1, 32, 0, stream>>>(acc, BATCH * 16);
  icp_final_acc_kernel<<<(BATCH * NPTS) / 256, 256, 0, stream>>>(psrc, pc, acc);
  icp_final_out_kernel<<<1, 64, 0, stream>>>(acc, out);
}